// Emb_37967510896970
// MI455X (gfx1250) — compile-verified
//
#include <hip/hip_runtime.h>

// Fourier embedding: out[e, i] = cos(pi * i * x[e]), e = 0..N-1, i = 0..127.
//
// Strategy (MI455X / gfx1250, wave32):
//  - Argument matrix is a rank-1 outer product (0.5*x) (x) i  -> V_WMMA_F32_16X16X4_F32.
//  - cos via hardware v_cos_f32 (computes cos(2*pi*s), so s = 0.5*i*x; |s| <= 63.5).
//  - 512 MB output >> 192 MB L2 and never re-read -> non-temporal stores.
//  - Each wave: 16 elements x 128 indices = 8 WMMA tiles, 64 v_cos, 64 NT b32 stores.

typedef float v2f __attribute__((ext_vector_type(2)));
typedef float v8f __attribute__((ext_vector_type(8)));

#define D_EMB 128

__global__ __launch_bounds__(256) void emb_wmma_kernel(const float* __restrict__ x,
                                                       float* __restrict__ out) {
  const int lane = threadIdx.x & 31;
  const int wave = threadIdx.x >> 5;                       // 0..7
  const size_t e0 = (size_t)blockIdx.x * 128u + (size_t)wave * 16u;

  // ---- A matrix (16x4 f32): column K=0 holds 0.5*x[e0+m]; columns 1..3 are zero.
  // Documented layout: lanes 0-15 carry M=0..15; VGPR0 = {K=0 | K=2}, VGPR1 = {K=1 | K=3}.
  float xv = 0.0f;
  if (lane < 16) xv = 0.5f * x[e0 + (size_t)lane];         // reconverges before WMMA
  v2f a;
  a.x = (lane < 16) ? xv : 0.0f;                           // K=0 (lanes 0-15), K=2 -> 0
  a.y = 0.0f;                                              // K=1 / K=3 -> 0

  // ---- Per-lane position in the D (16x16) tile:
  // VGPR r, lane l -> element m = r + 8*(l>=16), column n = l & 15.
  const int mhalf = (lane >> 4) << 3;                      // 0 or 8
  const int ncol  = lane & 15;
  float* obase = out + (e0 + (size_t)mhalf) * D_EMB + (size_t)ncol;

#pragma unroll
  for (int i0 = 0; i0 < D_EMB; i0 += 16) {
    // ---- B matrix (4x16 f32): row K=0 must hold the column indices (i0+n).
    // Rows K>0 are don't-care (A's columns 1..3 are zero), so replicate the
    // index pattern into every B slot: B[k, n] = i0 + n for all k.
    const float bval = (float)(i0 + ncol);
    v2f b;
    b.x = bval;
    b.y = bval;

    v8f c = {0.f, 0.f, 0.f, 0.f, 0.f, 0.f, 0.f, 0.f};
    // D[m, n] = 0.5 * x[e0+m] * (i0+n)   (normalized cos argument)
    v8f d = __builtin_amdgcn_wmma_f32_16x16x4_f32(
        /*neg_a=*/false, a, /*neg_b=*/false, b,
        /*c_mod=*/(short)0, c, /*reuse_a=*/false, /*reuse_b=*/false);

#pragma unroll
    for (int r = 0; r < 8; ++r) {
      const float v = __builtin_amdgcn_cosf(d[r]);         // cos(2*pi*s) = cos(pi*i*x)
      __builtin_nontemporal_store(v, obase + (size_t)r * D_EMB + (size_t)i0);
    }
  }
}

// Generic tail for element counts not divisible by 128 (not hit for 256x4096).
__global__ void emb_tail_kernel(const float* __restrict__ x, float* __restrict__ out,
                                long long e_start, long long e_end) {
  const long long t = (long long)blockIdx.x * blockDim.x + threadIdx.x;
  const long long e = e_start + t / D_EMB;
  const int i = (int)(t % D_EMB);
  if (e < e_end) {
    const float s = 0.5f * x[e] * (float)i;
    out[e * D_EMB + i] = __builtin_amdgcn_cosf(s);
  }
}

extern "C" void kernel_launch(void* const* d_in, const int* in_sizes, int n_in,
                              void* d_out, int out_size, void* d_ws, size_t ws_size,
                              hipStream_t stream) {
  const float* x = (const float*)d_in[0];
  float* out = (float*)d_out;

  const long long n_elems = (long long)in_sizes[0];        // 256*4096 = 1,048,576
  const long long groups = n_elems / 128;                  // elements per block = 128

  if (groups > 0) {
    emb_wmma_kernel<<<(unsigned)groups, 256, 0, stream>>>(x, out);
  }

  const long long rem = n_elems - groups * 128;
  if (rem > 0) {
    const long long work = rem * D_EMB;
    const int tb = 256;
    emb_tail_kernel<<<(unsigned)((work + tb - 1) / tb), tb, 0, stream>>>(
        x, out, groups * 128, n_elems);
  }
}